// GraphAttentionLayer_5119601017281
// MI455X (gfx1250) — compile-verified
//
#include <hip/hip_runtime.h>
#include <hip/hip_bf16.h>

// ---------------------------------------------------------------------------
// GAT layer, restructured:
//   h = x @ W^T                               (WMMA f32 16x16x4 GEMM)
//   hsum[c]   = sum_j h[j][c]                 (column sum, shared by all rows)
//   per edge: e  = leakyrelu(a . [h_s,h_d,ea])
//             w  = exp(e) - 1
//             acc[s][:] += w * h[d][:];  denom[s] += w
//   out[i][c] = (acc[i][c] + hsum[c]) / (denom[i] + N)
// which is exactly softmax over the dense N x N matrix with zeros in all
// non-edge slots (exp(0)=1 contributions factored out analytically).
// ---------------------------------------------------------------------------

#define GAT_N       8192
#define GAT_E       262144
#define GAT_NODEIN  128
#define GAT_EDGEIN  64
#define GAT_OUT     128

typedef float v2f __attribute__((ext_vector_type(2)));
typedef float v8f __attribute__((ext_vector_type(8)));

// ---------------------------------------------------------------------------
// Kernel 0: zero the accumulators (acc, denom, hsum)
// ---------------------------------------------------------------------------
__global__ void gat_zero_kernel(float* __restrict__ p, int n) {
    int i = blockIdx.x * blockDim.x + threadIdx.x;
    if (i < n) p[i] = 0.0f;
}

// ---------------------------------------------------------------------------
// Kernel 1: h = x @ W^T via V_WMMA_F32_16X16X4_F32.
// One wave (32 lanes) computes one 16x16 tile of h; 8 waves per block.
// A[m][k] = x[mtile*16+m][k], B[k][n] = W[ntile*16+n][k]   (W^T tile)
// Fragment layouts per CDNA5 ISA 7.12.2:
//   A (16x4 f32, 2 VGPRs): lanes 0-15 -> M=lane, (v0,v1)=(K0,K1)
//                          lanes 16-31 -> M=lane-16, (v0,v1)=(K2,K3)
//   B mirrors A's K split across lane halves; rows striped across lanes.
//   C/D (8 VGPRs): VGPR r, lanes 0-15 -> (M=r, N=lane); lanes 16-31 -> (M=r+8).
// ---------------------------------------------------------------------------
__global__ __launch_bounds__(256) void gat_gemm_h_kernel(
    const float* __restrict__ x,    // [N, 128]
    const float* __restrict__ W,    // [128, 128]
    float* __restrict__ h)          // [N, 128]
{
    const int wave = threadIdx.x >> 5;
    const int lane = threadIdx.x & 31;
    const int tile = blockIdx.x * 8 + wave;
    const int OT   = GAT_OUT / 16;              // 8 tiles along out-channels
    const int mi   = tile / OT;                 // node-row tile
    const int oi   = tile % OT;                 // out-channel tile

    const int aRow = mi * 16 + (lane & 15);     // M index for A loads
    const int bRow = oi * 16 + (lane & 15);     // N index -> row of W
    const int kHalf = (lane >> 4) * 2;          // 0 for lanes 0-15, 2 for 16-31

    const float* xA = x + (size_t)aRow * GAT_NODEIN;
    const float* wB = W + (size_t)bRow * GAT_NODEIN;

    v8f acc = {};
    #pragma unroll
    for (int k = 0; k < GAT_NODEIN; k += 4) {
        v2f a, b;
        a.x = xA[k + kHalf + 0];
        a.y = xA[k + kHalf + 1];
        b.x = wB[k + kHalf + 0];
        b.y = wB[k + kHalf + 1];
        acc = __builtin_amdgcn_wmma_f32_16x16x4_f32(
            /*neg_a=*/false, a, /*neg_b=*/false, b,
            /*c_mod=*/(short)0, acc, /*reuse_a=*/false, /*reuse_b=*/false);
    }

    const int n     = oi * 16 + (lane & 15);
    const int mBase = mi * 16 + (lane >> 4) * 8;
    #pragma unroll
    for (int r = 0; r < 8; ++r)
        h[(size_t)(mBase + r) * GAT_OUT + n] = acc[r];
}

// ---------------------------------------------------------------------------
// Kernel 2: hsum[c] = sum_j h[j][c].  128 threads/block, 64 rows/block,
// coalesced row-major reads, one atomic per (block, channel).
// ---------------------------------------------------------------------------
__global__ __launch_bounds__(128) void gat_colsum_kernel(
    const float* __restrict__ h, float* __restrict__ hsum)
{
    const int c  = threadIdx.x;
    const int r0 = blockIdx.x * 64;
    float s = 0.0f;
    #pragma unroll 8
    for (int r = 0; r < 64; ++r)
        s += h[(size_t)(r0 + r) * GAT_OUT + c];
    atomicAdd(&hsum[c], s);
}

// ---------------------------------------------------------------------------
// Kernel 3: one wave32 per edge.
//   e = leakyrelu( sum_c a[c]*h[s][c] + a[128+c]*h[d][c] + a[256+c]*ea[c] )
//   w = exp(e) - 1
//   denom[s] += w ;  acc[s][c] += w * h[d][c]
// ---------------------------------------------------------------------------
__global__ __launch_bounds__(256) void gat_edge_kernel(
    const int*   __restrict__ edge_index,  // [2, E] flattened
    const float* __restrict__ edge_attr,   // [E, 64]
    const float* __restrict__ a,           // [320]
    const float* __restrict__ h,           // [N, 128]
    float* __restrict__ acc,               // [N, 128]
    float* __restrict__ denom,             // [N]
    int E)
{
    const int lane = threadIdx.x & 31;
    const int e    = blockIdx.x * 8 + (threadIdx.x >> 5);
    if (e >= E) return;

    const int s = edge_index[e];
    const int d = edge_index[E + e];

    const float* hs = h + (size_t)s * GAT_OUT;
    const float* hd = h + (size_t)d * GAT_OUT;
    const float* ea = edge_attr + (size_t)e * GAT_EDGEIN;

    float p = 0.0f;
    #pragma unroll
    for (int c = lane; c < GAT_OUT; c += 32)
        p += a[c] * hs[c] + a[GAT_OUT + c] * hd[c];
    #pragma unroll
    for (int c = lane; c < GAT_EDGEIN; c += 32)
        p += a[2 * GAT_OUT + c] * ea[c];

    // wave32 reduction
    #pragma unroll
    for (int off = 16; off > 0; off >>= 1)
        p += __shfl_xor(p, off, 32);

    const float ev = (p > 0.0f) ? p : 0.2f * p;   // leaky_relu(0.2)
    const float w  = __expf(ev) - 1.0f;

    if (lane == 0) atomicAdd(&denom[s], w);
    float* as = acc + (size_t)s * GAT_OUT;
    #pragma unroll
    for (int c = lane; c < GAT_OUT; c += 32)
        atomicAdd(&as[c], w * hd[c]);
}

// ---------------------------------------------------------------------------
// Kernel 4: out[i][c] = (acc[i][c] + hsum[c]) / (denom[i] + N)
// ---------------------------------------------------------------------------
__global__ __launch_bounds__(128) void gat_finalize_kernel(
    const float* __restrict__ acc,
    const float* __restrict__ denom,
    const float* __restrict__ hsum,
    float* __restrict__ out)
{
    const int i = blockIdx.x;
    const int c = threadIdx.x;
    const float dn = denom[i] + (float)GAT_N;
    out[(size_t)i * GAT_OUT + c] = (acc[(size_t)i * GAT_OUT + c] + hsum[c]) / dn;
}

// ---------------------------------------------------------------------------
// Launcher. d_in order: x, edge_index, edge_attr, W, a.
// Workspace layout (floats): h[N*128] | acc[N*128] | denom[N] | hsum[128]
// ---------------------------------------------------------------------------
extern "C" void kernel_launch(void* const* d_in, const int* in_sizes, int n_in,
                              void* d_out, int out_size, void* d_ws, size_t ws_size,
                              hipStream_t stream) {
    const float* x          = (const float*)d_in[0];
    const int*   edge_index = (const int*)  d_in[1];
    const float* edge_attr  = (const float*)d_in[2];
    const float* W          = (const float*)d_in[3];
    const float* a          = (const float*)d_in[4];
    float*       out        = (float*)d_out;

    const int E = in_sizes[1] / 2;

    float* h     = (float*)d_ws;
    float* acc   = h + (size_t)GAT_N * GAT_OUT;
    float* denom = acc + (size_t)GAT_N * GAT_OUT;
    float* hsum  = denom + GAT_N;

    // zero acc | denom | hsum (contiguous)
    const int nzero = GAT_N * GAT_OUT + GAT_N + GAT_OUT;
    gat_zero_kernel<<<(nzero + 255) / 256, 256, 0, stream>>>(acc, nzero);

    // h = x @ W^T  (WMMA)
    const int ntiles = (GAT_N / 16) * (GAT_OUT / 16);   // 4096 waves
    gat_gemm_h_kernel<<<ntiles / 8, 256, 0, stream>>>(x, W, h);

    // column sum of h
    gat_colsum_kernel<<<GAT_N / 64, 128, 0, stream>>>(h, hsum);

    // per-edge attention scatter
    gat_edge_kernel<<<(E + 7) / 8, 256, 0, stream>>>(edge_index, edge_attr, a,
                                                     h, acc, denom, E);

    // normalize
    gat_finalize_kernel<<<GAT_N, 128, 0, stream>>>(acc, denom, hsum, out);
}